// Agent_14989435863205
// MI455X (gfx1250) — compile-verified
//
#include <hip/hip_runtime.h>

typedef __attribute__((ext_vector_type(2))) float v2f;
typedef __attribute__((ext_vector_type(8))) float v8f;

#define N_NODES 100000
#define N_EDGES 3200000

// ---------------------------------------------------------------- utilities
__global__ void zero_f32(float* __restrict__ p, long long n) {
  long long i = (long long)blockIdx.x * blockDim.x + threadIdx.x;
  long long stride = (long long)gridDim.x * blockDim.x;
  for (; i < n; i += stride) p[i] = 0.0f;
}

__global__ void edge_degree(const int* __restrict__ dst, float* __restrict__ deg, int E) {
  int e = blockIdx.x * blockDim.x + threadIdx.x;
  if (e < E) atomicAdd(&deg[dst[e]], 1.0f);
}

__global__ void deg_inv_sqrt_k(const float* __restrict__ deg, float* __restrict__ dis, int n) {
  int i = blockIdx.x * blockDim.x + threadIdx.x;
  if (i < n) dis[i] = rsqrtf(deg[i] + 1.0f);   // +1 for the self-loop
}

// ------------------------------------------------- fp32 WMMA GEMM (16x16 tile/wave)
// C[M,N] = A[M,K] @ B[K,N] (+bias, +relu). M,N multiples of 16, K multiple of 4.
// One wave32 per 16x16 tile; uses V_WMMA_F32_16X16X4_F32.
__global__ void gemm16_wmma_f32(const float* __restrict__ A, const float* __restrict__ B,
                                const float* __restrict__ bias, float* __restrict__ C,
                                int K, int Fout, int relu) {
  const int lane = threadIdx.x;        // 0..31
  const int half = lane >> 4;          // lane-half selects K sub-pair (ISA A/B layout)
  const int l    = lane & 15;
  const int rowBase = blockIdx.x * 16;
  const int colBase = blockIdx.y * 16;

  v8f acc = {};
  // A 16x4 f32 layout: lanes 0-15 -> {K=0,K=1}, lanes 16-31 -> {K=2,K=3}, M = lane&15
  const float* arow = A + (size_t)(rowBase + l) * K + 2 * half;
  // B 4x16 f32 layout (mirrors A): lanes 0-15 rows K=0/1, lanes 16-31 rows K=2/3, N = lane&15
  const float* bcol = B + (size_t)(2 * half) * Fout + colBase + l;

  for (int k0 = 0; k0 < K; k0 += 4) {
    v2f a, b;
    a.x = arow[k0];
    a.y = arow[k0 + 1];
    b.x = bcol[(size_t)k0 * Fout];
    b.y = bcol[(size_t)(k0 + 1) * Fout];
    acc = __builtin_amdgcn_wmma_f32_16x16x4_f32(false, a, false, b, (short)0, acc,
                                                false, false);
  }

  // C/D layout: VGPR r holds M = r (lanes 0-15) / M = 8+r (lanes 16-31), N = lane&15
#pragma unroll
  for (int r = 0; r < 8; ++r) {
    int row = rowBase + half * 8 + r;
    int col = colBase + l;
    float v = acc[r];
    if (bias) v += bias[col];
    if (relu) v = fmaxf(v, 0.0f);
    C[(size_t)row * Fout + col] = v;
  }
}

// ------------------------------------------------ edge message scatter-add
// thread = (edge, 4-feature chunk). Consecutive threads -> consecutive chunks of
// the same h[src] row => coalesced 128B gathers; atomics land in L2 (acc fits).
__global__ void edge_scatter(const int* __restrict__ src, const int* __restrict__ dst,
                             const float* __restrict__ dis, const float* __restrict__ h,
                             float* __restrict__ acc, int E, int F, int cshift) {
  long long tid = (long long)blockIdx.x * blockDim.x + threadIdx.x;
  long long total = (long long)E << cshift;
  if (tid >= total) return;
  int e = (int)(tid >> cshift);
  int c = ((int)tid & ((1 << cshift) - 1)) << 2;
  int s = src[e], d = dst[e];
  float w = dis[s] * dis[d];
  const float4 hv = *(const float4*)(h + (size_t)s * F + c);
  float* op = acc + (size_t)d * F + c;
  atomicAdd(op + 0, hv.x * w);
  atomicAdd(op + 1, hv.y * w);
  atomicAdd(op + 2, hv.z * w);
  atomicAdd(op + 3, hv.w * w);
}

// --------------------------------- self-loop + bias + ReLU, in place on acc
__global__ void finalize_k(float* __restrict__ acc, const float* __restrict__ h,
                           const float* __restrict__ dis, const float* __restrict__ bias,
                           long long total, int fshift) {
  long long i = (long long)blockIdx.x * blockDim.x + threadIdx.x;
  if (i >= total) return;
  int node = (int)(i >> fshift);
  int f = (int)i & ((1 << fshift) - 1);
  float dv = dis[node];
  float v = acc[i] + h[i] * dv * dv + bias[f];
  acc[i] = fmaxf(v, 0.0f);
}

// --------------------------------- final 64 -> 1 projection
__global__ void final_dense(const float* __restrict__ a3, const float* __restrict__ W,
                            const float* __restrict__ b, float* __restrict__ out, int N) {
  int i = blockIdx.x * blockDim.x + threadIdx.x;
  if (i >= N) return;
  const float4* av = (const float4*)(a3 + (size_t)i * 64);
  const float4* wv = (const float4*)W;
  float s = 0.0f;
#pragma unroll
  for (int j = 0; j < 16; ++j) {
    float4 a = av[j], w = wv[j];
    s += a.x * w.x + a.y * w.y + a.z * w.z + a.w * w.w;
  }
  out[i] = s + b[0];
}

// ---------------------------------------------------------------- launcher
extern "C" void kernel_launch(void* const* d_in, const int* in_sizes, int n_in,
                              void* d_out, int out_size, void* d_ws, size_t ws_size,
                              hipStream_t stream) {
  const float* x   = (const float*)d_in[0];
  const int*   ei  = (const int*)d_in[1];
  const float* W1  = (const float*)d_in[2];
  const float* b1  = (const float*)d_in[3];
  const float* W2  = (const float*)d_in[4];
  const float* b2  = (const float*)d_in[5];
  const float* Wl1 = (const float*)d_in[6];
  const float* bl1 = (const float*)d_in[7];
  const float* Wl2 = (const float*)d_in[8];
  const float* bl2 = (const float*)d_in[9];
  float* out = (float*)d_out;

  const int N = N_NODES;
  const int E = N_EDGES;
  const int* src = ei;
  const int* dst = ei + E;

  float* ws   = (float*)d_ws;
  float* deg  = ws;
  float* dis  = ws + N;
  float* bufA = ws + 2 * (size_t)N;            // N*128 floats
  float* bufB = bufA + (size_t)N * 128;        // N*128 floats

  // --- degrees & symmetric norm ---
  zero_f32<<<1024, 256, 0, stream>>>(deg, N);
  edge_degree<<<(E + 255) / 256, 256, 0, stream>>>(dst, deg, E);
  deg_inv_sqrt_k<<<(N + 255) / 256, 256, 0, stream>>>(deg, dis, N);

  // --- GCN layer 1: h1 = x @ W1  (24 -> 128) ---
  gemm16_wmma_f32<<<dim3(N / 16, 128 / 16), 32, 0, stream>>>(x, W1, nullptr, bufA, 24, 128, 0);
  zero_f32<<<4096, 256, 0, stream>>>(bufB, (long long)N * 128);
  {
    long long tot = (long long)E << 5;  // 32 chunks of 4 feats
    edge_scatter<<<(unsigned)((tot + 255) / 256), 256, 0, stream>>>(src, dst, dis, bufA,
                                                                    bufB, E, 128, 5);
    long long nf = (long long)N * 128;
    finalize_k<<<(unsigned)((nf + 255) / 256), 256, 0, stream>>>(bufB, bufA, dis, b1, nf, 7);
  }
  // a1 = bufB

  // --- GCN layer 2: h2 = a1 @ W2  (128 -> 64) ---
  gemm16_wmma_f32<<<dim3(N / 16, 64 / 16), 32, 0, stream>>>(bufB, W2, nullptr, bufA, 128, 64, 0);
  zero_f32<<<4096, 256, 0, stream>>>(bufB, (long long)N * 64);
  {
    long long tot = (long long)E << 4;  // 16 chunks of 4 feats
    edge_scatter<<<(unsigned)((tot + 255) / 256), 256, 0, stream>>>(src, dst, dis, bufA,
                                                                    bufB, E, 64, 4);
    long long nf = (long long)N * 64;
    finalize_k<<<(unsigned)((nf + 255) / 256), 256, 0, stream>>>(bufB, bufA, dis, b2, nf, 6);
  }
  // a2 = bufB

  // --- dense 1: a3 = relu(a2 @ Wl1 + bl1)  (64 -> 64) ---
  gemm16_wmma_f32<<<dim3(N / 16, 64 / 16), 32, 0, stream>>>(bufB, Wl1, bl1, bufA, 64, 64, 1);

  // --- dense 2: out = a3 @ Wl2 + bl2  (64 -> 1) ---
  final_dense<<<(N + 255) / 256, 256, 0, stream>>>(bufA, Wl2, bl2, out, N);
}